// HeteroGCNCiteer_7318624272994
// MI455X (gfx1250) — compile-verified
//
#include <hip/hip_runtime.h>
#include <hip/hip_bf16.h>

// ---------------------------------------------------------------------------
// HeteroGCN (2-layer, R=3 relations) for MI455X / gfx1250.
//   y_r = D_in^-1/2 A_r^T (D_out^-1/2 x) W_r + b_r,  summed over r, ReLU between
// GEMM via V_WMMA_F32_16X16X4_F32 (exact fp32 path); SpMM via fp32 global
// atomics with per-relation message buffers that fit in the 192MB L2.
// ---------------------------------------------------------------------------

typedef __attribute__((ext_vector_type(2))) float v2f;
typedef __attribute__((ext_vector_type(8))) float v8f;

static constexpr int Nn  = 100000;
static constexpr int Ee  = 1600000;
static constexpr int Rr  = 3;
static constexpr int IND = 128;
static constexpr int HID = 128;
static constexpr int OUTD = 64;

// ---------------- degree kernels ----------------

__global__ void zero_f32_kernel(float* __restrict__ p, int n) {
    int i = blockIdx.x * blockDim.x + threadIdx.x;
    if (i < n) p[i] = 0.0f;
}

__global__ void deg_count_kernel(const int* __restrict__ edges,
                                 float* __restrict__ deg_out,
                                 float* __restrict__ deg_in) {
    int e = blockIdx.x * blockDim.x + threadIdx.x;
    int r = blockIdx.y;
    if (e >= Ee) return;
    const int* base = edges + (size_t)r * 2 * Ee;
    int s = base[e];
    int d = base[Ee + e];
    unsafeAtomicAdd(&deg_out[r * Nn + s], 1.0f);
    unsafeAtomicAdd(&deg_in [r * Nn + d], 1.0f);
}

__global__ void rsqrt_deg_kernel(float* __restrict__ deg, int n) {
    int i = blockIdx.x * blockDim.x + threadIdx.x;
    if (i < n) deg[i] = rsqrtf(fmaxf(deg[i], 1.0f));
}

// ---------------- bias init: acc[n][c] = sum_r b[r][c] ----------------

template <int C>
__global__ void bias_init_kernel(const float* __restrict__ b, float* __restrict__ acc, int total) {
    int i = blockIdx.x * blockDim.x + threadIdx.x;
    if (i >= total) return;
    int c = i & (C - 1);
    acc[i] = b[c] + b[C + c] + b[2 * C + c];
}

// ---------------- WMMA GEMM:  M = (relu?(X)+scale) @ W  ----------------
// X: [N,128] fp32, scaled per-row by rs[]; W: [128,OUT_N]; M: [N,OUT_N].
// One block per 16-row tile; OUT_N/16 waves, one 16x16 WMMA tile per wave.

template <int OUT_N>
__global__ void gemm_wmma_kernel(const float* __restrict__ X,
                                 const float* __restrict__ rs,
                                 const float* __restrict__ W,
                                 float* __restrict__ M,
                                 int apply_relu) {
    __shared__ float lA[16 * 132];           // padded stride-132 to avoid bank conflicts

    const int tm      = blockIdx.x;          // row tile
    const int rowbase = tm * 16;
    const int lane    = threadIdx.x & 31;
    const int wave    = threadIdx.x >> 5;    // == column tile index tn

    // Cooperative load of the 16x128 A tile, scaled (and ReLU'd for layer 2).
    for (int j = threadIdx.x; j < 16 * 128; j += blockDim.x) {
        int rrow = j >> 7;
        int ccol = j & 127;
        float v = X[(size_t)(rowbase + rrow) * 128 + ccol];
        if (apply_relu) v = fmaxf(v, 0.0f);
        lA[rrow * 132 + ccol] = v * rs[rowbase + rrow];
    }
    __syncthreads();

    const int m  = lane & 15;   // A row / B col within tile
    const int kh = lane >> 4;   // which K-pair half this lane holds
    const int col = wave * 16 + m;

    v8f acc = {0.f, 0.f, 0.f, 0.f, 0.f, 0.f, 0.f, 0.f};

    #pragma unroll 4
    for (int k = 0; k < 128; k += 4) {
        const int k0 = k + kh * 2;
        v2f a, b;
        a.x = lA[m * 132 + k0];
        a.y = lA[m * 132 + k0 + 1];
        b.x = W[(size_t)k0 * OUT_N + col];
        b.y = W[(size_t)(k0 + 1) * OUT_N + col];
        acc = __builtin_amdgcn_wmma_f32_16x16x4_f32(
            /*neg_a=*/false, a, /*neg_b=*/false, b,
            /*c_mod=*/(short)0, acc, /*reuse_a=*/false, /*reuse_b=*/false);
    }

    // D layout: VGPR j -> (M=j, N=lane) for lanes 0-15, (M=j+8, N=lane-16) for 16-31.
    const int n    = lane & 15;
    const int half = lane >> 4;
    #pragma unroll
    for (int j = 0; j < 8; ++j) {
        int row = rowbase + half * 8 + j;
        M[(size_t)row * OUT_N + wave * 16 + n] = acc[j];
    }
}

// ---------------- edge scatter:  H[dst] += M[src] * rs_in[dst] ----------------
// C/4 lanes per edge, float4 coalesced gather, 4 hw fp32 atomics per lane.

template <int C>
__global__ void scatter_kernel(const float* __restrict__ M,
                               const int* __restrict__ src,
                               const int* __restrict__ dst,
                               const float* __restrict__ rs_in,
                               float* __restrict__ H) {
    constexpr int TPE = C / 4;               // threads per edge
    int t = blockIdx.x * blockDim.x + threadIdx.x;
    int e = t / TPE;
    if (e >= Ee) return;
    int cq = (t % TPE) * 4;

    int s = src[e];
    int d = dst[e];
    float sc = rs_in[d];
    const float4 v = *(const float4*)&M[(size_t)s * C + cq];
    float* out = &H[(size_t)d * C + cq];
    unsafeAtomicAdd(out + 0, v.x * sc);
    unsafeAtomicAdd(out + 1, v.y * sc);
    unsafeAtomicAdd(out + 2, v.z * sc);
    unsafeAtomicAdd(out + 3, v.w * sc);
}

// ---------------- launch ----------------

extern "C" void kernel_launch(void* const* d_in, const int* in_sizes, int n_in,
                              void* d_out, int out_size, void* d_ws, size_t ws_size,
                              hipStream_t stream) {
    const float* x     = (const float*)d_in[0];   // [N,128]
    const int*   edges = (const int*)  d_in[1];   // [R,2,E]
    const float* W1    = (const float*)d_in[2];   // [R,128,128]
    const float* b1    = (const float*)d_in[3];   // [R,128]
    const float* W2    = (const float*)d_in[4];   // [R,128,64]
    const float* b2    = (const float*)d_in[5];   // [R,64]
    float*       out   = (float*)d_out;           // [N,64]

    // Workspace layout (floats):
    //   deg/rs : 2*R*N   (deg_out then deg_in; rsqrt'd in place)
    //   h      : N*128   (layer-1 accumulator)
    //   m      : N*128   (per-relation message buffer, reused both layers)
    float* wsf    = (float*)d_ws;
    float* rs_out = wsf;                       // [R*N]
    float* rs_in  = wsf + (size_t)Rr * Nn;     // [R*N]
    float* h      = wsf + (size_t)2 * Rr * Nn; // [N*128]
    float* m      = h + (size_t)Nn * HID;      // [N*128]

    const int B = 256;

    // Degrees.
    int degN = 2 * Rr * Nn;
    zero_f32_kernel<<<(degN + B - 1) / B, B, 0, stream>>>(wsf, degN);
    deg_count_kernel<<<dim3((Ee + B - 1) / B, Rr), B, 0, stream>>>(edges, rs_out, rs_in);
    rsqrt_deg_kernel<<<(degN + B - 1) / B, B, 0, stream>>>(wsf, degN);

    // Bias init of both accumulators.
    bias_init_kernel<HID><<<((Nn * HID) + B - 1) / B, B, 0, stream>>>(b1, h, Nn * HID);
    bias_init_kernel<OUTD><<<((Nn * OUTD) + B - 1) / B, B, 0, stream>>>(b2, out, Nn * OUTD);

    // Layer 1: per relation, GEMM then scatter (stream order serializes reuse of m).
    for (int r = 0; r < Rr; ++r) {
        const int* src = edges + (size_t)r * 2 * Ee;
        const int* dst = src + Ee;
        gemm_wmma_kernel<HID><<<Nn / 16, (HID / 16) * 32, 0, stream>>>(
            x, rs_out + (size_t)r * Nn, W1 + (size_t)r * IND * HID, m, /*relu=*/0);
        scatter_kernel<HID><<<(Ee * (HID / 4) + B - 1) / B, B, 0, stream>>>(
            m, src, dst, rs_in + (size_t)r * Nn, h);
    }

    // Layer 2: ReLU fused into the GEMM A-load.
    for (int r = 0; r < Rr; ++r) {
        const int* src = edges + (size_t)r * 2 * Ee;
        const int* dst = src + Ee;
        gemm_wmma_kernel<OUTD><<<Nn / 16, (OUTD / 16) * 32, 0, stream>>>(
            h, rs_out + (size_t)r * Nn, W2 + (size_t)r * HID * OUTD, m, /*relu=*/1);
        scatter_kernel<OUTD><<<(Ee * (OUTD / 4) + B - 1) / B, B, 0, stream>>>(
            m, src, dst, rs_in + (size_t)r * Nn, out);
    }
}